// SGRU_89696097009971
// MI455X (gfx1250) — compile-verified
//
#include <hip/hip_runtime.h>

typedef float v2f __attribute__((ext_vector_type(2)));
typedef float v8f __attribute__((ext_vector_type(8)));

__device__ __forceinline__ float sigmoidf(float x) {
    return 1.0f / (1.0f + __expf(-x));
}

// C[M,N] = A[M,K] @ W[N,K]^T + bias[N], optional relu.
// One wave (32 threads) computes one 16x16 output tile via V_WMMA_F32_16X16X4_F32.
// A-frag layout (ISA 7.12.2, 32-bit A 16x4): lanes 0-15 hold M=lane, K=k0..k0+1;
// lanes 16-31 hold M=lane-16, K=k0+2..k0+3.  B-frag mirrors with N in place of M.
// C/D layout: VGPR i, lanes 0-15 -> (M=i, N=lane); lanes 16-31 -> (M=i+8, N=lane-16).
__global__ __launch_bounds__(32)
void wmma_gemm_nt(const float* __restrict__ A, const float* __restrict__ W,
                  const float* __restrict__ bias, float* __restrict__ C,
                  int M, int N, int K, int relu)
{
    const int lane = threadIdx.x;
    const int half = lane >> 4;       // 0 or 1
    const int rr   = lane & 15;
    const int n0 = blockIdx.x * 16;
    const int m0 = blockIdx.y * 16;

    const float* __restrict__ arow = A + (size_t)(m0 + rr) * K;
    const float* __restrict__ wrow = W + (size_t)(n0 + rr) * K;

    v8f acc = {0.f, 0.f, 0.f, 0.f, 0.f, 0.f, 0.f, 0.f};
    for (int k = 0; k < K; k += 4) {
        const int ka = k + (half << 1);
        v2f a, b;
        a[0] = arow[ka];     a[1] = arow[ka + 1];
        b[0] = wrow[ka];     b[1] = wrow[ka + 1];
        acc = __builtin_amdgcn_wmma_f32_16x16x4_f32(
            /*neg_a=*/false, a, /*neg_b=*/false, b,
            /*c_mod=*/(short)0, acc, /*reuse_a=*/false, /*reuse_b=*/false);
    }

    const float bn = bias ? bias[n0 + rr] : 0.0f;
#pragma unroll
    for (int i = 0; i < 8; ++i) {
        float val = acc[i] + bn;
        if (relu) val = fmaxf(val, 0.0f);
        C[(size_t)(m0 + i + (half << 3)) * N + (n0 + rr)] = val;
    }
}

// plastic[b,i] = |alpha[i]| * dot(dU[b,i,:], h[b,:]);  H=512, one wave per row.
__global__ __launch_bounds__(256)
void plastic_kernel(const float* __restrict__ dU, const float* __restrict__ h,
                    const float* __restrict__ alpha, float* __restrict__ out)
{
    const int wave = threadIdx.x >> 5;
    const int lane = threadIdx.x & 31;
    const int row  = blockIdx.x * 8 + wave;   // 0 .. B*H-1
    const int b = row >> 9;
    const int i = row & 511;

    const float4* __restrict__ du = (const float4*)(dU + (size_t)row * 512);
    const float4* __restrict__ hb = (const float4*)(h  + (size_t)b   * 512);

    float s = 0.0f;
#pragma unroll
    for (int t = 0; t < 4; ++t) {
        float4 d  = du[lane + t * 32];
        float4 hv = hb[lane + t * 32];
        s += d.x * hv.x + d.y * hv.y + d.z * hv.z + d.w * hv.w;
    }
#pragma unroll
    for (int off = 16; off > 0; off >>= 1)
        s += __shfl_xor(s, off, 32);

    if (lane == 0) out[row] = fabsf(alpha[i]) * s;
}

// Per-element [B,H] cell math: gates, new_v, swish(new_h), new_tr_e.
// Also emits s-gate and g = nh*(nh - ne) for the big trace pass.
__global__ __launch_bounds__(256)
void cell_pointwise(const float* __restrict__ Wx, const float* __restrict__ Wh,
                    const float* __restrict__ plastic,
                    const float* __restrict__ v_in, const float* __restrict__ e_in,
                    float* __restrict__ v_out, float* __restrict__ h_out,
                    float* __restrict__ e_out, float* __restrict__ s_out,
                    float* __restrict__ g_out)
{
    const int idx = blockIdx.x * blockDim.x + threadIdx.x;   // B*H total
    const int b = idx >> 9;
    const int i = idx & 511;
    const size_t base = (size_t)b * 2048;

    const float z = sigmoidf(Wx[base + i]        + Wh[base + i]);
    const float r = sigmoidf(Wx[base + 512 + i]  + Wh[base + 512 + i]);
    const float s = sigmoidf(Wx[base + 1024 + i] + Wh[base + 1024 + i]);
    const float dv = Wx[base + 1536 + i] + r * (Wh[base + 1536 + i] + plastic[idx]);
    const float nv = (1.0f - z) * v_in[idx] + z * dv;
    const float nh = nv * sigmoidf(nv);
    const float ne = (1.0f - z) * e_in[idx] + z * nh * nh;

    v_out[idx] = nv;
    h_out[idx] = nh;
    e_out[idx] = ne;
    s_out[idx] = s;
    g_out[idx] = nh * (nh - ne);
}

// Streaming [B,H,H] pass: one 128-thread block per (b,i) row (512 floats = 128 float4).
//   outer = g[b,i] * h_prev[b,j]
//   nE    = (1-s)*trE + s*outer
//   ndU   = clip(dU + sigmoid(tau[i])*m[b,i]*nE, -2, 2)
__global__ __launch_bounds__(128)
void trace_update(const float* __restrict__ dU_in, const float* __restrict__ trE_in,
                  const float* __restrict__ s, const float* __restrict__ g,
                  const float* __restrict__ m, const float* __restrict__ tau,
                  const float* __restrict__ h_prev,
                  float* __restrict__ dU_out, float* __restrict__ trE_out)
{
    const int row = blockIdx.x;           // 0 .. B*H-1
    const int b = row >> 9;
    const int i = row & 511;
    const size_t off = (size_t)row * 512 + (size_t)threadIdx.x * 4;

    float4 d  = *(const float4*)(dU_in  + off);
    float4 E  = *(const float4*)(trE_in + off);
    float4 hv = *(const float4*)(h_prev + (size_t)b * 512 + (size_t)threadIdx.x * 4);

    const float si   = s[row];
    const float gi   = g[row];
    const float coef = sigmoidf(tau[i]) * m[row];
    const float os   = 1.0f - si;

    float4 nE, nD;
    nE.x = os * E.x + si * (gi * hv.x);
    nE.y = os * E.y + si * (gi * hv.y);
    nE.z = os * E.z + si * (gi * hv.z);
    nE.w = os * E.w + si * (gi * hv.w);
    nD.x = fminf(2.0f, fmaxf(-2.0f, d.x + coef * nE.x));
    nD.y = fminf(2.0f, fmaxf(-2.0f, d.y + coef * nE.y));
    nD.z = fminf(2.0f, fmaxf(-2.0f, d.z + coef * nE.z));
    nD.w = fminf(2.0f, fmaxf(-2.0f, d.w + coef * nE.w));

    *(float4*)(trE_out + off) = nE;
    *(float4*)(dU_out  + off) = nD;
}

extern "C" void kernel_launch(void* const* d_in, const int* in_sizes, int n_in,
                              void* d_out, int out_size, void* d_ws, size_t ws_size,
                              hipStream_t stream)
{
    const int L = 2, B = 64, H = 512, I = 512, R = 32, O = 512;
    const int H4 = 4 * H;
    const size_t BH  = (size_t)B * H;
    const size_t BHH = BH * H;

    const float* x     = (const float*)d_in[0];
    const float* h_in  = (const float*)d_in[1];
    const float* v_in  = (const float*)d_in[2];
    const float* dU_in = (const float*)d_in[3];
    const float* te_in = (const float*)d_in[4];
    const float* tE_in = (const float*)d_in[5];
    const float* x2hW  = (const float*)d_in[6];
    const float* x2hb  = (const float*)d_in[7];
    const float* h2hW  = (const float*)d_in[8];
    const float* h2hb  = (const float*)d_in[9];
    const float* alpha = (const float*)d_in[10];
    const float* tauU  = (const float*)d_in[11];
    const float* mW1   = (const float*)d_in[12];
    const float* mb1   = (const float*)d_in[13];
    const float* mW2   = (const float*)d_in[14];
    const float* mb2   = (const float*)d_in[15];
    const float* decW  = (const float*)d_in[16];
    const float* decb  = (const float*)d_in[17];

    // Output tuple, concatenated flat: vs, hs, dUs, tes, tEs, output
    float* out = (float*)d_out;
    float* vs  = out;
    float* hs  = out + (size_t)L * BH;
    float* dUs = out + (size_t)2 * L * BH;
    float* tes = out + (size_t)2 * L * BH + (size_t)L * BHH;
    float* tEs = out + (size_t)3 * L * BH + (size_t)L * BHH;
    float* dec = out + (size_t)3 * L * BH + (size_t)2 * L * BHH;

    // Workspace carve-up (~1.6 MB)
    float* ws     = (float*)d_ws;
    float* Wx     = ws;  ws += (size_t)B * H4;
    float* Wh     = ws;  ws += (size_t)B * H4;
    float* plast  = ws;  ws += BH;
    float* s_ws   = ws;  ws += BH;
    float* g_ws   = ws;  ws += BH;
    float* modhid = ws;  ws += (size_t)B * R;
    float* m_ws   = ws;  ws += BH;

    for (int l = 0; l < L; ++l) {
        const float* inp = (l == 0) ? x : (hs + (size_t)(l - 1) * BH);
        const float* hl  = h_in + (size_t)l * BH;

        // Wx = inp @ x2h_W[l].T + x2h_b[l]   ;   Wh = h[l] @ h2h_W[l].T + h2h_b[l]
        wmma_gemm_nt<<<dim3(H4 / 16, B / 16), 32, 0, stream>>>(
            inp, x2hW + (size_t)l * H4 * I, x2hb + (size_t)l * H4, Wx, B, H4, I, 0);
        wmma_gemm_nt<<<dim3(H4 / 16, B / 16), 32, 0, stream>>>(
            hl, h2hW + (size_t)l * H4 * H, h2hb + (size_t)l * H4, Wh, B, H4, H, 0);

        // plastic[b,i] = |alpha_i| * dot(dU[l,b,i,:], h[l,b,:])
        plastic_kernel<<<(B * H) / 8, 256, 0, stream>>>(
            dU_in + (size_t)l * BHH, hl, alpha + (size_t)l * H, plast);

        // gates / new_v / new_h / new_tr_e (+ s and g side outputs)
        cell_pointwise<<<(B * H) / 256, 256, 0, stream>>>(
            Wx, Wh, plast, v_in + (size_t)l * BH, te_in + (size_t)l * BH,
            vs + (size_t)l * BH, hs + (size_t)l * BH, tes + (size_t)l * BH,
            s_ws, g_ws);

        // m = relu(nh @ mW1.T + mb1) @ mW2.T + mb2
        wmma_gemm_nt<<<dim3(R / 16, B / 16), 32, 0, stream>>>(
            hs + (size_t)l * BH, mW1 + (size_t)l * R * H, mb1 + (size_t)l * R,
            modhid, B, R, H, 1);
        wmma_gemm_nt<<<dim3(H / 16, B / 16), 32, 0, stream>>>(
            modhid, mW2 + (size_t)l * H * R, mb2 + (size_t)l * H, m_ws, B, H, R, 0);

        // big streaming pass over [B,H,H]: new trace_E and clipped new dU
        trace_update<<<B * H, 128, 0, stream>>>(
            dU_in + (size_t)l * BHH, tE_in + (size_t)l * BHH, s_ws, g_ws, m_ws,
            tauU + (size_t)l * H, hl,
            dUs + (size_t)l * BHH, tEs + (size_t)l * BHH);
    }

    // output = h_last @ dec_W.T + dec_b
    wmma_gemm_nt<<<dim3(O / 16, B / 16), 32, 0, stream>>>(
        hs + (size_t)(L - 1) * BH, decW, decb, dec, B, O, H, 0);
}